// CurveAttention_17179869184496
// MI455X (gfx1250) — compile-verified
//
#include <hip/hip_runtime.h>
#include <math.h>

typedef __attribute__((ext_vector_type(2))) float v2f;
typedef __attribute__((ext_vector_type(8))) float v8f;

#define SEQ 2048
#define HD  64
#define KSTRIDE 68   // 16x64 tile padded: row stride 68 floats -> conflict-free b64 gathers, rows 16B-aligned
#define WSTRIDE 20   // 16x16 weight tile padded: conflict-free A-fragment reads

__global__ __launch_bounds__(256)
void curve_attention_kernel(const float* __restrict__ Q,
                            const float* __restrict__ K,
                            const float* __restrict__ V,
                            float* __restrict__ out,   // [B*H, S, D]
                            float* __restrict__ wts) { // [B*H, S, S]
    __shared__ __align__(16) float lK[16 * KSTRIDE];
    __shared__ __align__(16) float lV[16 * KSTRIDE];
    __shared__ __align__(16) float lW[8][16 * WSTRIDE];

    const int tid  = threadIdx.x;
    const int wave = tid >> 5;
    const int lane = tid & 31;
    const int hi   = lane >> 4;     // half-wave select
    const int ln   = lane & 15;
    const int kdb  = hi * 2;        // K-dim base held by this half-wave (A/B frags)

    const int bh    = blockIdx.x >> 4;   // 0..63  (B*H)
    const int qblk  = blockIdx.x & 15;   // 0..15
    const int qbase = qblk * 128 + wave * 16;

    const float* Qb = Q + (size_t)bh * SEQ * HD;
    const float* Kb = K + (size_t)bh * SEQ * HD;
    const float* Vb = V + (size_t)bh * SEQ * HD;
    float* Ob = out + (size_t)bh * SEQ * HD;
    float* Wb = wts + (size_t)bh * SEQ * (size_t)SEQ;

    const float scale = 0.125f;                      // D^-0.5, folded into Q fragments
    const float LOG2E = 1.44269504088896340736f;     // log2(e)
    const float LOG2C = 4.01384324854532f;           // log2(1 + e^e)

    // Preload Q A-fragments (pre-scaled): a_q[kc] covers K-dims kc*4 + {kdb,kdb+1}, row = qbase+ln.
    v2f a_q[16];
    {
        const float* qrow = Qb + (size_t)(qbase + ln) * HD + kdb;
#pragma unroll
        for (int kc = 0; kc < 16; ++kc) {
            v2f q = *(const v2f*)(qrow + kc * 4);
            a_q[kc].x = q.x * scale;
            a_q[kc].y = q.y * scale;
        }
    }

    v8f acc[4];
#pragma unroll
    for (int ct = 0; ct < 4; ++ct)
#pragma unroll
        for (int r = 0; r < 8; ++r) acc[ct][r] = 0.0f;

    for (int j = 0; j < SEQ / 16; ++j) {
        // Cooperative staging: 16x64 K and V tiles -> LDS (coalesced b128).
        {
            const int r  = tid >> 4;          // 0..15 tile row
            const int c4 = (tid & 15) * 4;    // 0..60
            const float4 kk = *(const float4*)(Kb + (size_t)(j * 16 + r) * HD + c4);
            const float4 vv = *(const float4*)(Vb + (size_t)(j * 16 + r) * HD + c4);
            *(float4*)&lK[r * KSTRIDE + c4] = kk;
            *(float4*)&lV[r * KSTRIDE + c4] = vv;
        }
        __syncthreads();

        // scores tile = (Q*scale)(16xD) * K^T(Dx16), D chunked by 4: 16x v_wmma_f32_16x16x4_f32
        v8f c;
#pragma unroll
        for (int r = 0; r < 8; ++r) c[r] = 0.0f;
#pragma unroll
        for (int kc = 0; kc < 16; ++kc) {
            // B(kd,n) = K[j*16+n][kc*4+kd]; lane holds n=ln, kd = kdb,kdb+1
            v2f b = *(const v2f*)&lK[ln * KSTRIDE + kc * 4 + kdb];
            c = __builtin_amdgcn_wmma_f32_16x16x4_f32(false, a_q[kc], false, b,
                                                      (short)0, c, false, false);
        }

        // softemax: exp(s)/(1+e^e) = 2^(s*log2e - log2C)
        // raw v_exp_f32 (no libm subnormal guard: |arg| << 126 for this data,
        // and underflow-to-zero matches the reference's exp anyway)
        float w[8];
#pragma unroll
        for (int r = 0; r < 8; ++r)
            w[r] = __builtin_amdgcn_exp2f(fmaf(c[r], LOG2E, -LOG2C));

        // attn_weights (output #2): streaming, write-once -> non-temporal stores
        // (keeps K/V working set resident in L2; weights never re-read)
#pragma unroll
        for (int r = 0; r < 8; ++r)
            __builtin_nontemporal_store(w[r],
                &Wb[(size_t)(qbase + r + hi * 8) * SEQ + (j * 16 + ln)]);

        // Stage weight tile in per-wave LDS to re-swizzle C-layout -> A-layout.
        float* wt = lW[wave];
#pragma unroll
        for (int r = 0; r < 8; ++r)
            wt[(r + hi * 8) * WSTRIDE + ln] = w[r];
        __builtin_amdgcn_wave_barrier();   // keep DS store->load ordered (per-wave tile)

        // out(16x64) += W(16x16) * V(16x64), key dim chunked by 4
#pragma unroll
        for (int kc = 0; kc < 4; ++kc) {
            // A(m,kd) = W[m][kc*4+kd]; lane holds m=ln, kd = kdb,kdb+1
            v2f aW = *(const v2f*)&wt[ln * WSTRIDE + kc * 4 + kdb];
#pragma unroll
            for (int ct = 0; ct < 4; ++ct) {
                // B(kd,n) = V[j*16 + kc*4+kd][ct*16+n]
                v2f bV;
                bV.x = lV[(kc * 4 + kdb)     * KSTRIDE + ct * 16 + ln];
                bV.y = lV[(kc * 4 + kdb + 1) * KSTRIDE + ct * 16 + ln];
                acc[ct] = __builtin_amdgcn_wmma_f32_16x16x4_f32(false, aW, false, bV,
                                                                (short)0, acc[ct], false, false);
            }
        }
        __syncthreads();   // protect lK/lV before next iteration's staging
    }

    // attn_output (output #1): C-layout store, coalesced per half-wave
#pragma unroll
    for (int ct = 0; ct < 4; ++ct)
#pragma unroll
        for (int r = 0; r < 8; ++r)
            Ob[(size_t)(qbase + r + hi * 8) * HD + ct * 16 + ln] = acc[ct][r];
}

extern "C" void kernel_launch(void* const* d_in, const int* in_sizes, int n_in,
                              void* d_out, int out_size, void* d_ws, size_t ws_size,
                              hipStream_t stream) {
    (void)in_sizes; (void)n_in; (void)d_ws; (void)ws_size; (void)out_size;
    const float* Q = (const float*)d_in[0];
    const float* K = (const float*)d_in[1];
    const float* V = (const float*)d_in[2];
    float* out = (float*)d_out;                                   // [4,16,2048,64]
    float* wts = out + (size_t)4 * 16 * 2048 * 64;                // [4,16,2048,2048]

    // grid: B*H (64) * (S/128 = 16) q-row blocks; 8 waves/block, 16 q-rows/wave
    dim3 grid(64 * 16), block(256);
    curve_attention_kernel<<<grid, block, 0, stream>>>(Q, K, V, out, wts);
}